// HeteroGraphSAGE_37245956391038
// MI455X (gfx1250) — compile-verified
//
#include <hip/hip_runtime.h>

typedef float v2f __attribute__((ext_vector_type(2)));
typedef float v8f __attribute__((ext_vector_type(8)));

#define NFEAT 64

// -------------------- utility kernels --------------------

__global__ void zero_kernel(float* __restrict__ p, size_t n) {
  size_t i = (size_t)blockIdx.x * blockDim.x + threadIdx.x;
  size_t stride = (size_t)gridDim.x * blockDim.x;
  for (; i < n; i += stride) p[i] = 0.0f;
}

// o = 0.5*(a+b), elementwise (weight / bias pre-combination)
__global__ void combine_avg_kernel(const float* __restrict__ a,
                                   const float* __restrict__ b,
                                   float* __restrict__ o, int n) {
  int i = blockIdx.x * blockDim.x + threadIdx.x;
  if (i < n) o[i] = 0.5f * (a[i] + b[i]);
}

__global__ void degree_kernel(const int* __restrict__ dst, float* __restrict__ cnt, int E) {
  int e = blockIdx.x * blockDim.x + threadIdx.x;
  if (e < E) atomicAdd(&cnt[dst[e]], 1.0f);
}

// agg[dst[e]] += y[src[e]]  (64 floats/edge, 16 lanes per edge, float4 each)
__global__ void scatter_add_kernel(const float* __restrict__ y,
                                   const int* __restrict__ src,
                                   const int* __restrict__ dst,
                                   float* __restrict__ agg, int E) {
  long gid = (long)blockIdx.x * blockDim.x + threadIdx.x;
  int e = (int)(gid >> 4);
  if (e >= E) return;
  int q = (int)(gid & 15);
  int s = src[e], d = dst[e];
  const float4 v = ((const float4*)(y + (size_t)s * NFEAT))[q];
  float* o = agg + (size_t)d * NFEAT + q * 4;
  atomicAdd(o + 0, v.x);
  atomicAdd(o + 1, v.y);
  atomicAdd(o + 2, v.z);
  atomicAdd(o + 3, v.w);
}

// out = aggA/max(cntA,1) + aggB/max(cntB,1)   (y already carries the 0.5 and Wl)
__global__ void mean_combine_kernel(const float* __restrict__ aggA, const float* __restrict__ cntA,
                                    const float* __restrict__ aggB, const float* __restrict__ cntB,
                                    float* __restrict__ out, int total) {
  int i = blockIdx.x * blockDim.x + threadIdx.x;
  if (i >= total) return;
  int n = i >> 6;
  out[i] = aggA[i] / fmaxf(cntA[n], 1.0f) + aggB[i] / fmaxf(cntB[n], 1.0f);
}

// -------------------- WMMA GEMM: C[M x 64] (+)= scale*(A[M x K] @ W[K x 64]) + bias --------------------
// M % 16 == 0 (holds for 100000/50000/10000), K in {64,128}.
// Block = 128 threads = 4 waves; block -> one 16-row tile of A; wave w -> output cols [16w,16w+16).
__global__ void __launch_bounds__(128)
gemm_n64_wmma(const float* __restrict__ A, const float* __restrict__ W,
              const float* __restrict__ bias, float* __restrict__ C,
              int K, float scale, int accumulate, int relu) {
  __shared__ float tileA[16 * 132];            // K<=128, padded stride K+4 -> conflict-free ds_load_b64
  const int ldk = K + 4;
  const size_t rowBase = (size_t)blockIdx.x * 16;

  for (int t = threadIdx.x; t < 16 * K; t += 128) {      // coalesced global -> LDS stage
    int r = t / K, c = t - r * K;
    tileA[r * ldk + c] = A[(rowBase + r) * (size_t)K + c];
  }
  __syncthreads();

  const int lane = threadIdx.x & 31;
  const int wv   = threadIdx.x >> 5;           // n-tile 0..3
  const int half = lane >> 4;                  // 0: lanes 0-15, 1: lanes 16-31
  const int l16  = lane & 15;
  const int col  = wv * 16 + l16;

  v8f acc = {0.f, 0.f, 0.f, 0.f, 0.f, 0.f, 0.f, 0.f};
  const float* ap = &tileA[l16 * ldk + 2 * half];        // A: row l16, K offset 2*half (+k)
  const float* wp = &W[(size_t)(2 * half) * NFEAT + col]; // B: row 2*half (+k), col

  for (int k = 0; k < K; k += 4) {
    v2f a, b;
    a.x = ap[k];
    a.y = ap[k + 1];
    b.x = wp[(size_t)k * NFEAT];
    b.y = wp[(size_t)(k + 1) * NFEAT];
    acc = __builtin_amdgcn_wmma_f32_16x16x4_f32(false, a, false, b, (short)0, acc, false, false);
  }

  const float bc = bias ? bias[col] : 0.0f;
#pragma unroll
  for (int i = 0; i < 8; ++i) {                // acc[i] -> row i + 8*half (C/D layout)
    size_t r = rowBase + (size_t)(i + 8 * half);
    float v = acc[i] * scale + bc;
    size_t idx = r * NFEAT + col;
    if (accumulate) v += C[idx];
    if (relu) v = fmaxf(v, 0.0f);
    C[idx] = v;
  }
}

// -------------------- row-wise L2 normalize (one wave per 64-float row) --------------------
__global__ void __launch_bounds__(128)
l2norm_kernel(float* __restrict__ x, int nrows) {
  int wv = threadIdx.x >> 5;
  int lane = threadIdx.x & 31;
  int row = blockIdx.x * 4 + wv;
  if (row >= nrows) return;
  float* p = x + (size_t)row * NFEAT;
  float a = p[lane], b = p[lane + 32];
  float s = a * a + b * b;
  for (int m = 16; m >= 1; m >>= 1) s += __shfl_xor(s, m, 32);
  float inv = 1.0f / fmaxf(sqrtf(s), 1e-12f);
  p[lane] = a * inv;
  p[lane + 32] = b * inv;
}

// -------------------- orchestration --------------------

static inline dim3 g1(long n, int b) { return dim3((unsigned)((n + b - 1) / b)); }

extern "C" void kernel_launch(void* const* d_in, const int* in_sizes, int n_in,
                              void* d_out, int out_size, void* d_ws, size_t ws_size,
                              hipStream_t stream) {
  const float* xc = (const float*)d_in[0];
  const float* xm = (const float*)d_in[1];
  const float* xd = (const float*)d_in[2];
  const int NC = in_sizes[0] / 128;
  const int NM = in_sizes[1] / 128;
  const int ND = in_sizes[2] / 128;
  const int E  = in_sizes[3];

  const int* srcE[6] = {(const int*)d_in[3],  (const int*)d_in[5],  (const int*)d_in[7],
                        (const int*)d_in[9],  (const int*)d_in[11], (const int*)d_in[13]};
  const int* dstE[6] = {(const int*)d_in[4],  (const int*)d_in[6],  (const int*)d_in[8],
                        (const int*)d_in[10], (const int*)d_in[12], (const int*)d_in[14]};
  const float* W1l = (const float*)d_in[15];
  const float* W1r = (const float*)d_in[16];
  const float* b1  = (const float*)d_in[17];
  const float* W2l = (const float*)d_in[18];
  const float* W2r = (const float*)d_in[19];
  const float* b2  = (const float*)d_in[20];
  const float* Wres = (const float*)d_in[21];
  const float* bres = (const float*)d_in[22];

  // relation r: 0=c->m 1=m->d 2=c->d 3=m->c 4=d->m 5=d->c ; node types 0=c 1=m 2=d
  const int rsrc[6] = {0, 1, 0, 1, 2, 2};
  const int rdst[6] = {1, 2, 2, 0, 1, 0};
  const int Nnode[3] = {NC, NM, ND};
  const float* Xin[3] = {xc, xm, xd};
  // per dst type, which two relations feed it: c:(mc,dc) m:(cm,dm) d:(md,cd)
  const int pairA[3] = {3, 0, 1};
  const int pairB[3] = {5, 4, 2};

  // ---- workspace layout ----
  float* ws = (float*)d_ws;
  size_t off = 0;
  float* cnt[6];
  for (int r = 0; r < 6; ++r) { cnt[r] = ws + off; off += (size_t)Nnode[rdst[r]]; }
  size_t aggStart = off;
  float* agg[6];
  for (int r = 0; r < 6; ++r) { agg[r] = ws + off; off += (size_t)Nnode[rdst[r]] * NFEAT; }
  size_t aggTotal = off - aggStart;
  size_t cntAggTotal = off;
  int maxN = NC > NM ? NC : NM; if (ND > maxN) maxN = ND;
  float* ybuf = ws + off; off += (size_t)maxN * NFEAT;
  float* h[3];
  for (int t = 0; t < 3; ++t) { h[t] = ws + off; off += (size_t)Nnode[t] * NFEAT; }
  float* Wc1 = ws + off; off += 3 * 128 * 64;
  float* bc1 = ws + off; off += 3 * 64;
  float* Wc2 = ws + off; off += 3 * 64 * 64;
  float* bc2 = ws + off; off += 3 * 64;
  (void)ws_size; (void)n_in; (void)out_size;

  // ---- phase 0: zero counters+accumulators, pre-combine self weights, degrees ----
  zero_kernel<<<dim3(2048), dim3(256), 0, stream>>>(ws, cntAggTotal);
  for (int t = 0; t < 3; ++t) {
    combine_avg_kernel<<<g1(128 * 64, 256), dim3(256), 0, stream>>>(
        W1r + (size_t)pairA[t] * 128 * 64, W1r + (size_t)pairB[t] * 128 * 64, Wc1 + (size_t)t * 128 * 64, 128 * 64);
    combine_avg_kernel<<<g1(64, 64), dim3(64), 0, stream>>>(
        b1 + pairA[t] * 64, b1 + pairB[t] * 64, bc1 + t * 64, 64);
    combine_avg_kernel<<<g1(64 * 64, 256), dim3(256), 0, stream>>>(
        W2r + (size_t)pairA[t] * 64 * 64, W2r + (size_t)pairB[t] * 64 * 64, Wc2 + (size_t)t * 64 * 64, 64 * 64);
    combine_avg_kernel<<<g1(64, 64), dim3(64), 0, stream>>>(
        b2 + pairA[t] * 64, b2 + pairB[t] * 64, bc2 + t * 64, 64);
  }
  for (int r = 0; r < 6; ++r)
    degree_kernel<<<g1(E, 256), dim3(256), 0, stream>>>(dstE[r], cnt[r], E);

  // ---- layer 1: y = x_src @ (0.5*W1_l[r]) ; scatter-mean ; h = relu(sum + x@Wc1 + bc1) ----
  for (int r = 0; r < 6; ++r) {
    int s = rsrc[r];
    gemm_n64_wmma<<<dim3(Nnode[s] / 16), dim3(128), 0, stream>>>(
        Xin[s], W1l + (size_t)r * 128 * 64, nullptr, ybuf, 128, 0.5f, 0, 0);
    scatter_add_kernel<<<g1((long)E * 16, 256), dim3(256), 0, stream>>>(
        ybuf, srcE[r], dstE[r], agg[r], E);
  }
  for (int t = 0; t < 3; ++t) {
    int rA = pairA[t], rB = pairB[t];
    mean_combine_kernel<<<g1((long)Nnode[t] * NFEAT, 256), dim3(256), 0, stream>>>(
        agg[rA], cnt[rA], agg[rB], cnt[rB], h[t], Nnode[t] * NFEAT);
    gemm_n64_wmma<<<dim3(Nnode[t] / 16), dim3(128), 0, stream>>>(
        Xin[t], Wc1 + (size_t)t * 128 * 64, bc1 + t * 64, h[t], 128, 1.0f, 1, 1);
  }

  // ---- layer 2 ----
  zero_kernel<<<dim3(2048), dim3(256), 0, stream>>>(ws + aggStart, aggTotal);
  for (int r = 0; r < 6; ++r) {
    int s = rsrc[r];
    gemm_n64_wmma<<<dim3(Nnode[s] / 16), dim3(128), 0, stream>>>(
        h[s], W2l + (size_t)r * 64 * 64, nullptr, ybuf, 64, 0.5f, 0, 0);
    scatter_add_kernel<<<g1((long)E * 16, 256), dim3(256), 0, stream>>>(
        ybuf, srcE[r], dstE[r], agg[r], E);
  }

  float* outp = (float*)d_out;
  float* o[3] = {outp, outp + (size_t)NC * NFEAT, outp + (size_t)(NC + NM) * NFEAT};
  for (int t = 0; t < 3; ++t) {
    int rA = pairA[t], rB = pairB[t];
    mean_combine_kernel<<<g1((long)Nnode[t] * NFEAT, 256), dim3(256), 0, stream>>>(
        agg[rA], cnt[rA], agg[rB], cnt[rB], o[t], Nnode[t] * NFEAT);
    gemm_n64_wmma<<<dim3(Nnode[t] / 16), dim3(128), 0, stream>>>(
        h[t], Wc2 + (size_t)t * 64 * 64, bc2 + t * 64, o[t], 64, 1.0f, 1, 0);
    gemm_n64_wmma<<<dim3(Nnode[t] / 16), dim3(128), 0, stream>>>(
        Xin[t], Wres + (size_t)t * 128 * 64, bres + t * 64, o[t], 128, 1.0f, 1, 0);
    l2norm_kernel<<<g1(Nnode[t], 4), dim3(128), 0, stream>>>(o[t], Nnode[t]);
  }
}